// LocalSpatioTemporalPooling_59863254171814
// MI455X (gfx1250) — compile-verified
//
#include <hip/hip_runtime.h>
#include <hip/hip_bf16.h>

typedef __attribute__((ext_vector_type(2))) float v2f;
typedef __attribute__((ext_vector_type(8))) float v8f;

#define NN 8
#define CC 256
#define TT 30
#define HH 64
#define WW 44
#define SS 8
#define SEG_LEN 352            // (HH/SS)*WW contiguous floats per (n,c,t,stripe)
#define NUM_SEG (NN*CC*TT*SS)  // 491520
#define EPSV 1e-6f

// Stage 1: segmented sum of 352-float contiguous segments via V_WMMA_F32_16X16X4_F32.
// One wave reduces 16 segments. A-matrix layout (ISA 7.12.2, 32-bit A 16x4):
//   lanes 0-15 : VGPR0=K0, VGPR1=K1 for row M=lane
//   lanes 16-31: VGPR0=K2, VGPR1=K3 for row M=lane-16
// B = all ones -> every column of D is the row sum (layout independent).
// D layout: VGPR j: lanes 0-15 -> M=j, lanes 16-31 -> M=8+j.
__global__ void __launch_bounds__(256)
lstp_feats_wmma_kernel(const float* __restrict__ x, float* __restrict__ feats)
{
    const int tid    = threadIdx.x;
    const int lane   = tid & 31;
    const int waveId = (blockIdx.x * blockDim.x + tid) >> 5;
    const int segBase = waveId * 16;          // 16 segments per wave

    const int row = lane & 15;                // which of the 16 segments
    const int kh  = (lane >> 4) << 1;         // 0 for lanes 0-15, 2 for lanes 16-31

    const float* __restrict__ segPtr = x + (size_t)(segBase + row) * SEG_LEN + kh;

    v2f b; b[0] = 1.0f; b[1] = 1.0f;          // all-ones B matrix (both VGPRs)
    v8f acc = {};                              // C/D accumulator, 16x16 f32

    // 352 / 4 = 88 WMMA steps; each lane feeds a contiguous 8-byte pair.
    #pragma unroll 4
    for (int k = 0; k < SEG_LEN; k += 4) {
        v2f a = *(const v2f*)(segPtr + k);    // 8B aligned: seg*1408 + even*4
        acc = __builtin_amdgcn_wmma_f32_16x16x4_f32(
            /*neg_a=*/false, a, /*neg_b=*/false, b,
            /*c_mod=*/(short)0, acc, /*reuse_a=*/false, /*reuse_b=*/false);
    }

    // Lanes 0 and 16 each hold 8 row sums (N=0 column). Scatter to feats[n][s][t][c].
    if ((lane & 15) == 0) {
        const int half = lane >> 4;           // 0 -> rows 0-7, 1 -> rows 8-15
        const float inv = 1.0f / (float)SEG_LEN;
        #pragma unroll
        for (int j = 0; j < 8; ++j) {
            int seg = segBase + half * 8 + j; // flat (n,c,t,s) index
            int s   = seg & (SS - 1);
            int r   = seg >> 3;
            int t   = r % TT;  r /= TT;
            int c   = r & (CC - 1);
            int n   = r >> 8;
            feats[(((size_t)(n * SS + s) * TT) + t) * CC + c] = acc[j] * inv;
        }
    }
}

// Stage 2: one block per (n, stripe). feats slice is [TT][CC] contiguous.
// Deterministic score reduction, top-2 over time, mean of 2 frames.
__global__ void __launch_bounds__(256)
lstp_topk_pool_kernel(const float* __restrict__ feats, float* __restrict__ out)
{
    __shared__ float part[TT][8];
    __shared__ float sc[TT];
    __shared__ int   sel[2];

    const int tid = threadIdx.x;
    const size_t base = (size_t)blockIdx.x * TT * CC;

    // 240 threads: (t, p) partial sums of 32 channels each (fixed order -> deterministic)
    if (tid < TT * 8) {
        const int t = tid >> 3;
        const int p = tid & 7;
        const float* f = feats + base + (size_t)t * CC + p * 32;
        float s = 0.0f;
        #pragma unroll
        for (int i = 0; i < 32; ++i) { float v = f[i]; s += v * v; }
        part[t][p] = s;
    }
    __syncthreads();

    if (tid < TT) {
        float s = 0.0f;
        #pragma unroll
        for (int p = 0; p < 8; ++p) s += part[tid][p];
        sc[tid] = fmaxf(s, EPSV);   // clip; sqrt is monotone so skip it for ranking
    }
    __syncthreads();

    if (tid == 0) {
        int t0 = 0; float b0 = sc[0];
        for (int t = 1; t < TT; ++t) { if (sc[t] > b0) { b0 = sc[t]; t0 = t; } }
        int t1 = (t0 == 0) ? 1 : 0; float b1 = sc[t1];
        for (int t = 0; t < TT; ++t) {
            if (t != t0 && sc[t] > b1) { b1 = sc[t]; t1 = t; }
        }
        sel[0] = t0; sel[1] = t1;
    }
    __syncthreads();

    const int t0 = sel[0], t1 = sel[1];
    // out is (n, S*C) stripe-major == blockIdx.x * C + c
    out[(size_t)blockIdx.x * CC + tid] =
        0.5f * (feats[base + (size_t)t0 * CC + tid] +
                feats[base + (size_t)t1 * CC + tid]);
}

extern "C" void kernel_launch(void* const* d_in, const int* in_sizes, int n_in,
                              void* d_out, int out_size, void* d_ws, size_t ws_size,
                              hipStream_t stream)
{
    const float* x   = (const float*)d_in[0];
    float* feats     = (float*)d_ws;       // 491,520 floats ~ 1.97 MB
    float* out       = (float*)d_out;      // 8 x 2048 fp32

    // 16 segments/wave, 8 waves/block -> 128 segments/block; 491520/128 = 3840 blocks
    lstp_feats_wmma_kernel<<<NUM_SEG / 128, 256, 0, stream>>>(x, feats);
    // one block per (n, stripe) = 64 blocks
    lstp_topk_pool_kernel<<<NN * SS, 256, 0, stream>>>(feats, out);
}